// CannyLoss_16776142258463
// MI455X (gfx1250) — compile-verified
//
#include <hip/hip_runtime.h>
#include <math.h>

#define BATCH 32
#define H 512
#define W 512
#define HW (H * W)
#define NPIX (BATCH * HW)

typedef __attribute__((ext_vector_type(2))) float v2f;
typedef __attribute__((ext_vector_type(8))) float v8f;

// xor-16 lane exchange via DS_SWIZZLE_B32 group-of-32 mode (SWAPX16):
// offset = {xor=0x10, or=0x00, and=0x1f} -> 0x401F. Cheaper than the
// generic __shfl_xor lowering (no mbcnt/cndmask validity scaffolding).
__device__ __forceinline__ float swap16_add(float v) {
  int j = __builtin_amdgcn_ds_swizzle(__float_as_int(v), 0x401F);
  return v + __int_as_float(j);
}

// Wave-level sum using the CDNA5 matrix engine: V_WMMA_F32_16X16X4_F32.
// A (16x4 f32): VGPR0 holds K=0 (lanes 0-15) / K=2 (lanes 16-31); VGPR1 = 0.
// B (4x16) = all ones -> D[m][n] = s(lane m) + s(lane m+16) for every n.
// Per-lane sum of the 8 D registers covers half the rows; swap16 combines.
// Requires EXEC all ones (call sites are fully uniform, full waves).
__device__ __forceinline__ float wave_sum_wmma(float s) {
  v2f a; a.x = s;   a.y = 0.0f;
  v2f b; b.x = 1.0f; b.y = 1.0f;
  v8f c = {0.f, 0.f, 0.f, 0.f, 0.f, 0.f, 0.f, 0.f};
  c = __builtin_amdgcn_wmma_f32_16x16x4_f32(false, a, false, b, (short)0, c,
                                            false, false);
  float r = c[0] + c[1] + c[2] + c[3] + c[4] + c[5] + c[6] + c[7];
  return swap16_add(r);  // every lane holds the full 32-lane sum
}

// Pass 1: quantize labels, Sobel with replicate border, L1 magnitude +
// direction bin, packed into u16 (mag in [0,2040] -> 12 bits, bin -> 2 bits).
__global__ void k_sobel(const float* __restrict__ labels,
                        unsigned short* __restrict__ magbin) {
  int pix = blockIdx.x * blockDim.x + threadIdx.x;
  int b = pix / HW;
  int r = pix - b * HW;
  int y = r / W;
  int x = r - y * W;
  const float* L = labels + (long)b * HW;
  int ym = max(y - 1, 0), yp = min(y + 1, H - 1);
  int xm = max(x - 1, 0), xp = min(x + 1, W - 1);
  auto Q = [&](int yy, int xx) -> float {
    float v = L[yy * W + xx];
    return floorf(fminf(fmaxf(v, 0.0f), 1.0f) * 255.0f);
  };
  float i00 = Q(ym, xm), i01 = Q(ym, x), i02 = Q(ym, xp);
  float i10 = Q(y, xm),                  i12 = Q(y, xp);
  float i20 = Q(yp, xm), i21 = Q(yp, x), i22 = Q(yp, xp);
  float gx = (i02 + 2.0f * i12 + i22) - (i00 + 2.0f * i10 + i20);
  float gy = (i20 + 2.0f * i21 + i22) - (i00 + 2.0f * i01 + i02);
  float ax = fabsf(gx), ay = fabsf(gy);
  int mag = (int)(ax + ay);  // exact: integers <= 2040
  int bin;
  if (ax == 0.0f && ay == 0.0f) bin = 0;                      // atan2(0,0)=0
  else if (ay < 0.41421356237309503f * ax) bin = 0;           // theta<22.5 | >=157.5
  else if (ay > 2.4142135623730951f * ax) bin = 2;            // 67.5..112.5
  else bin = (gx * gy > 0.0f) ? 1 : 3;                        // 45 : 135
  magbin[pix] = (unsigned short)(mag | (bin << 12));
}

// Pass 2: NMS (zero-padded shifts) + double threshold.
// state: 0 = none, 1 = weak candidate, 2 = edge (strong).
__global__ void k_nms(const unsigned short* __restrict__ magbin,
                      unsigned char* __restrict__ state) {
  int pix = blockIdx.x * blockDim.x + threadIdx.x;
  int b = pix / HW;
  int r = pix - b * HW;
  int y = r / W;
  int x = r - y * W;
  const unsigned short* M = magbin + (long)b * HW;
  int p = M[y * W + x];
  int mag = p & 0xFFF;
  int bin = p >> 12;
  auto nb = [&](int dy, int dx) -> int {
    int yy = y + dy, xx = x + dx;
    if ((unsigned)yy >= (unsigned)H || (unsigned)xx >= (unsigned)W) return 0;
    return M[yy * W + xx] & 0xFFF;
  };
  int n1, n2;
  if (bin == 0)      { n1 = nb(0, 1);   n2 = nb(0, -1); }
  else if (bin == 1) { n1 = nb(-1, 1);  n2 = nb(1, -1); }
  else if (bin == 2) { n1 = nb(1, 0);   n2 = nb(-1, 0); }
  else               { n1 = nb(-1, -1); n2 = nb(1, 1);  }
  unsigned char s = 0;
  if (mag >= n1 && mag >= n2) {
    if (mag > 200) s = 2;
    else if (mag > 100) s = 1;
  }
  state[pix] = s;
}

// Pass 3 (repeated): hysteresis flood fill. Monotone 1->2 upgrades only, so
// in-place chaotic iteration (stale halos, racy stores of identical value 2)
// converges to the same unique least fixpoint as the reference while_loop.
#define TS 64
#define SM_STRIDE (TS + 8)
__global__ void k_hyst(unsigned char* __restrict__ state) {
  __shared__ unsigned char sm[(TS + 2) * SM_STRIDE];
  __shared__ int changed;
  __shared__ int anyChange;
  int x0 = blockIdx.x * TS, y0 = blockIdx.y * TS;
  unsigned char* S = state + (long)blockIdx.z * HW;
  int tid = threadIdx.y * blockDim.x + threadIdx.x;  // blockDim (64,4)

  for (int i = tid; i < (TS + 2) * (TS + 2); i += 256) {
    int ly = i / (TS + 2);
    int lx = i - ly * (TS + 2);
    int gy = y0 - 1 + ly, gx = x0 - 1 + lx;
    unsigned char v = 0;
    if ((unsigned)gy < (unsigned)H && (unsigned)gx < (unsigned)W)
      v = S[gy * W + gx];
    sm[ly * SM_STRIDE + lx] = v;
  }
  if (tid == 0) anyChange = 0;

  for (int it = 0; it < 96; ++it) {
    __syncthreads();            // all threads done reading `changed`
    if (tid == 0) changed = 0;
    __syncthreads();
    int lx = threadIdx.x + 1;
    for (int k = 0; k < 16; ++k) {
      int ly = threadIdx.y + 4 * k + 1;
      int idx = ly * SM_STRIDE + lx;
      if (sm[idx] == 1) {
        if (sm[idx - SM_STRIDE - 1] == 2 || sm[idx - SM_STRIDE] == 2 ||
            sm[idx - SM_STRIDE + 1] == 2 || sm[idx - 1] == 2 ||
            sm[idx + 1] == 2 || sm[idx + SM_STRIDE - 1] == 2 ||
            sm[idx + SM_STRIDE] == 2 || sm[idx + SM_STRIDE + 1] == 2) {
          sm[idx] = 2;
          changed = 1;
          anyChange = 1;
        }
      }
    }
    __syncthreads();
    if (changed == 0) break;    // uniform: read after barrier
  }

  if (anyChange) {              // skip writeback for already-converged tiles
    int lx = threadIdx.x + 1;
    for (int k = 0; k < 16; ++k) {
      int ly = threadIdx.y + 4 * k + 1;
      S[(y0 + ly - 1) * W + (x0 + lx - 1)] = sm[ly * SM_STRIDE + lx];
    }
  }
}

__device__ __forceinline__ float nll1(float p0, float p1, bool c) {
  float m = fmaxf(p0, p1);
  float lse = m + log1pf(expf(-fabsf(p0 - p1)));
  return lse - (c ? p1 : p0);
}

// Pass 4: NLL partial sums, float4-vectorized on the dominant 64 MiB stream.
// 2,097,152 groups / 524,288 threads = 4 uniform iterations -> EXEC all-1s
// at the WMMA reduction. Prefetch one stride ahead (global_prefetch_b8).
#define NBLK4 2048
__global__ void k_loss(const float* __restrict__ pred,
                       const unsigned char* __restrict__ state,
                       float* __restrict__ partials) {
  __shared__ float wsum[8];
  const int NG = NPIX / 4;
  int tid = blockIdx.x * blockDim.x + threadIdx.x;
  int nthreads = gridDim.x * blockDim.x;
  const unsigned int* st32 = (const unsigned int*)state;
  float s = 0.0f;
  for (int g = tid; g < NG; g += nthreads) {
    int b = g / (HW / 4);
    int r4 = g - b * (HW / 4);
    const float4* P0 = (const float4*)(pred + (long)b * 2 * HW);
    const float4* P1 = (const float4*)(pred + (long)b * 2 * HW + HW);
    // hint next stride; speculative prefetch past the end is dropped
    __builtin_prefetch((const char*)&P0[r4] + (long)nthreads * 16, 0, 3);
    __builtin_prefetch((const char*)&P1[r4] + (long)nthreads * 16, 0, 3);
    float4 a0 = P0[r4];
    float4 a1 = P1[r4];
    unsigned int sv = st32[g];
    s += nll1(a0.x, a1.x, (sv & 0xFFu) == 2u);
    s += nll1(a0.y, a1.y, ((sv >> 8) & 0xFFu) == 2u);
    s += nll1(a0.z, a1.z, ((sv >> 16) & 0xFFu) == 2u);
    s += nll1(a0.w, a1.w, (sv >> 24) == 2u);
  }
  float r32 = wave_sum_wmma(s);
  int lane = threadIdx.x & 31, wv = threadIdx.x >> 5;
  if (lane == 0) wsum[wv] = r32;
  __syncthreads();
  if (threadIdx.x == 0) {
    float t = 0.0f;
    for (int i = 0; i < 8; ++i) t += wsum[i];
    partials[blockIdx.x] = t;
  }
}

// Pass 5: deterministic final reduction + mean.
__global__ void k_final(const float* __restrict__ partials,
                        float* __restrict__ out) {
  __shared__ float wsum[8];
  float s = 0.0f;
  for (int i = threadIdx.x; i < NBLK4; i += 256) s += partials[i];
  float r = wave_sum_wmma(s);
  int lane = threadIdx.x & 31, wv = threadIdx.x >> 5;
  if (lane == 0) wsum[wv] = r;
  __syncthreads();
  if (threadIdx.x == 0) {
    float t = 0.0f;
    for (int i = 0; i < 8; ++i) t += wsum[i];
    out[0] = t * (1.0f / (float)NPIX);  // exact /2^23
  }
}

extern "C" void kernel_launch(void* const* d_in, const int* in_sizes, int n_in,
                              void* d_out, int out_size, void* d_ws,
                              size_t ws_size, hipStream_t stream) {
  (void)in_sizes; (void)n_in; (void)out_size; (void)ws_size;
  const float* pred = (const float*)d_in[0];     // [32,2,512,512] f32
  const float* labels = (const float*)d_in[1];   // [32,512,512] f32
  float* out = (float*)d_out;

  char* ws = (char*)d_ws;
  unsigned short* magbin = (unsigned short*)ws;                 // 16 MiB
  unsigned char* state = (unsigned char*)(ws + 2L * NPIX);      // 8 MiB
  float* partials = (float*)(ws + 3L * NPIX);                   // 8 KiB

  k_sobel<<<NPIX / 256, 256, 0, stream>>>(labels, magbin);
  k_nms<<<NPIX / 256, 256, 0, stream>>>(magbin, state);
  dim3 hgrid(W / TS, H / TS, BATCH);   // 8 x 8 x 32
  dim3 hblk(64, 4);
  for (int it = 0; it < 12; ++it)
    k_hyst<<<hgrid, hblk, 0, stream>>>(state);
  k_loss<<<NBLK4, 256, 0, stream>>>(pred, state, partials);
  k_final<<<1, 256, 0, stream>>>(partials, out);
}